// RippleConnectionGenerator_75247827026717
// MI455X (gfx1250) — compile-verified
//
#include <hip/hip_runtime.h>

typedef __attribute__((ext_vector_type(2))) float v2f;
typedef __attribute__((ext_vector_type(4))) float v4f;
typedef __attribute__((ext_vector_type(8))) float v8f;

#define K_TOP   10000
#define RB      512          // reduction blocks
#define TIE_CAP 128

// speed2 bits: must be bit-identical in every kernel that classifies a node.
__device__ __forceinline__ unsigned spd_bits(const float* __restrict__ nf, int i) {
  float vx = nf[i * 12 + 0], vy = nf[i * 12 + 1], vz = nf[i * 12 + 2];
  float s = __fadd_rn(__fadd_rn(__fmul_rn(vx, vx), __fmul_rn(vy, vy)), __fmul_rn(vz, vz));
  return __float_as_uint(s);   // speed2 >= 0 -> raw bits are order-isomorphic to value
}

// ---------------- workspace zeroing ----------------
__global__ void k_zero_u32(unsigned* p, int n) {
  int i = blockIdx.x * blockDim.x + threadIdx.x;
  const int st = gridDim.x * blockDim.x;
  for (; i < n; i += st) p[i] = 0u;
}

// ---------------- radix select, pass 0: top 16 bits ----------------
__global__ void k_hist0(const float* __restrict__ nf, int N, unsigned* __restrict__ hist) {
  int i = blockIdx.x * blockDim.x + threadIdx.x;
  const int st = gridDim.x * blockDim.x;
  for (; i < N; i += st) atomicAdd(&hist[spd_bits(nf, i) >> 16], 1u);
}

// ---------------- radix select, pass 1: low 16 bits within selected bucket ----------------
__global__ void k_hist1(const float* __restrict__ nf, int N,
                        const unsigned* __restrict__ state, unsigned* __restrict__ hist) {
  const unsigned pre = state[0];
  int i = blockIdx.x * blockDim.x + threadIdx.x;
  const int st = gridDim.x * blockDim.x;
  for (; i < N; i += st) {
    unsigned b = spd_bits(nf, i);
    if ((b >> 16) == pre) atomicAdd(&hist[b & 0xFFFFu], 1u);
  }
}

// descending cumulative scan over 65536 bins to locate the k-th largest bucket
__global__ __launch_bounds__(256) void k_scan(const unsigned* __restrict__ hist,
                                              unsigned* state, int pass) {
  __shared__ unsigned chunkSum[256];
  __shared__ unsigned binVal[256];
  __shared__ int selChunkS;
  __shared__ unsigned cumS;
  const int t = threadIdx.x;
  const unsigned kk = (pass == 0) ? (unsigned)K_TOP : state[1];
  unsigned s = 0;
  for (int j = 0; j < 256; ++j) s += hist[t * 256 + j];
  chunkSum[t] = s;
  __syncthreads();
  if (t == 0) {
    unsigned cum = 0; int sel = 0;
    for (int c = 255; c >= 0; --c) {
      unsigned cs = chunkSum[c];
      if (cum + cs >= kk) { sel = c; break; }
      cum += cs;
    }
    selChunkS = sel; cumS = cum;
  }
  __syncthreads();
  binVal[t] = hist[selChunkS * 256 + t];
  __syncthreads();
  if (t == 0) {
    unsigned cum = cumS; int sb = 0;
    for (int b = 255; b >= 0; --b) {
      unsigned h = binVal[b];
      if (cum + h >= kk) { sb = b; break; }
      cum += h;
    }
    unsigned sel = (unsigned)(selChunkS * 256 + sb);
    if (pass == 0) { state[0] = sel; state[1] = kk - cum; }
    else           { state[2] = (state[0] << 16) | sel; state[3] = kk - cum; }
  }
}

// ---------------- deterministic member reduction (f64 tree, no float atomics) ----------------
__global__ __launch_bounds__(256) void k_reduce(const float* __restrict__ nf,
                                                const float* __restrict__ wp,
                                                const float* __restrict__ mp, int N,
                                                const unsigned* __restrict__ state,
                                                double* __restrict__ partials,
                                                unsigned* tiectr, int* tielist) {
  __shared__ double sm[256][8];
  const unsigned T = state[2];
  double acc[8];
#pragma unroll
  for (int c = 0; c < 8; ++c) acc[c] = 0.0;
  int i = blockIdx.x * blockDim.x + threadIdx.x;
  const int st = gridDim.x * blockDim.x;
  for (; i < N; i += st) {
    unsigned b = spd_bits(nf, i);
    if (b > T) {
      acc[0] += nf[i * 12 + 0]; acc[1] += nf[i * 12 + 1]; acc[2] += nf[i * 12 + 2];
      acc[3] += wp[i * 3 + 0];  acc[4] += wp[i * 3 + 1];  acc[5] += wp[i * 3 + 2];
      acc[6] += mp[i * 2 + 0];  acc[7] += mp[i * 2 + 1];
    } else if (b == T) {
      unsigned p = atomicAdd(tiectr, 1u);
      if (p < TIE_CAP) tielist[p] = i;
    }
  }
  const int tid = threadIdx.x;
#pragma unroll
  for (int c = 0; c < 8; ++c) sm[tid][c] = acc[c];
  __syncthreads();
  for (int s = 128; s > 0; s >>= 1) {
    if (tid < s) {
#pragma unroll
      for (int c = 0; c < 8; ++c) sm[tid][c] += sm[tid + s][c];
    }
    __syncthreads();
  }
  if (tid == 0) {
#pragma unroll
    for (int c = 0; c < 8; ++c) partials[blockIdx.x * 8 + c] = sm[0][c];
  }
}

// ---------------- combine partials, resolve ties, build layer-normed x (64x20 padded) ----------------
__global__ __launch_bounds__(64) void k_finalize(const float* __restrict__ nf,
                                                 const float* __restrict__ wp,
                                                 const float* __restrict__ mp,
                                                 const int* __restrict__ ripple, int S,
                                                 const float* __restrict__ ln0s,
                                                 const float* __restrict__ ln0b,
                                                 const unsigned* __restrict__ state,
                                                 const double* __restrict__ partials,
                                                 const unsigned* tiectr, const int* tielist,
                                                 float* __restrict__ xbuf) {
  __shared__ double hv[8];
  __shared__ int list[TIE_CAP];
  const int t = threadIdx.x;
  if (t < 8) {
    double s = 0.0;
    for (int b = 0; b < RB; ++b) s += partials[b * 8 + t];   // fixed order: deterministic
    hv[t] = s;
  }
  __syncthreads();
  if (t == 0) {
    int need = (int)state[3];
    int n = (int)(*tiectr); if (n > TIE_CAP) n = TIE_CAP;
    for (int j = 0; j < n; ++j) list[j] = tielist[j];
    for (int a = 1; a < n; ++a) {                            // sort ascending: lowest-index ties win
      int v = list[a]; int b = a - 1;
      while (b >= 0 && list[b] > v) { list[b + 1] = list[b]; --b; }
      list[b + 1] = v;
    }
    int take = need < n ? need : n;
    for (int j = 0; j < take; ++j) {
      int i = list[j];
      hv[0] += nf[i * 12 + 0]; hv[1] += nf[i * 12 + 1]; hv[2] += nf[i * 12 + 2];
      hv[3] += wp[i * 3 + 0];  hv[4] += wp[i * 3 + 1];  hv[5] += wp[i * 3 + 2];
      hv[6] += mp[i * 2 + 0];  hv[7] += mp[i * 2 + 1];
    }
  }
  __syncthreads();
  if (t < 64) {
    float row[20];
#pragma unroll
    for (int j = 0; j < 20; ++j) row[j] = 0.f;
    if (t < S) {
      float h0 = (float)hv[0], h1 = (float)hv[1], h2 = (float)hv[2];
      float w0 = (float)hv[3], w1v = (float)hv[4], w2v = (float)hv[5];
      float m0 = (float)hv[6], m1 = (float)hv[7];
      int idx = ripple[t];
      row[0] = h0; row[1] = h1; row[2] = h2;
      row[3] = h0; row[4] = h1; row[5] = h2;
      row[6] = h0; row[7] = h1; row[8] = h2;
      row[9]  = nf[idx * 12 + 0]; row[10] = nf[idx * 12 + 1]; row[11] = nf[idx * 12 + 2];
      row[12] = wp[idx * 3 + 0] - w0;  row[13] = wp[idx * 3 + 1] - w1v; row[14] = wp[idx * 3 + 2] - w2v;
      row[15] = mp[idx * 2 + 0] - m0;  row[16] = mp[idx * 2 + 1] - m1;
      float mu = 0.f;
      for (int j = 0; j < 17; ++j) mu += row[j];
      mu *= (1.f / 17.f);
      float var = 0.f;
      for (int j = 0; j < 17; ++j) { float d = row[j] - mu; var += d * d; }
      var *= (1.f / 17.f);
      float inv = rsqrtf(var + 1e-5f);
      for (int j = 0; j < 17; ++j) row[j] = (row[j] - mu) * inv * ln0s[j] + ln0b[j];
      row[17] = row[18] = row[19] = 0.f;
    }
    for (int j = 0; j < 20; ++j) xbuf[t * 20 + j] = row[j];
  }
}

// ---------------- the roofline: 512 MB streaming copy latent -> out ----------------
__global__ void k_copy(const v4f* __restrict__ src, v4f* __restrict__ dst, int n4) {
  int i = blockIdx.x * blockDim.x + threadIdx.x;
  const int st = gridDim.x * blockDim.x;
  for (; i < n4; i += st) {
    v4f v = __builtin_nontemporal_load(&src[i]);
    __builtin_nontemporal_store(v, &dst[i]);
  }
}

// ---------------- WMMA f32 16x16x4 GEMM tile: one wave owns 64(M) x 16(N) ----------------
__device__ __forceinline__ v8f wmma4(v2f a, v2f b, v8f c) {
  return __builtin_amdgcn_wmma_f32_16x16x4_f32(false, a, false, b, (short)0, c, false, false);
}

__device__ __forceinline__ void gemm_tile(const float* As, int lda, int Ktot, int Kvalid,
                                          const float* __restrict__ Bg,
                                          const float* __restrict__ bias,
                                          float* Ds, bool relu, int lane, int n0) {
  v8f acc[4] = {};
  const int col  = lane & 15;             // N within tile (lanes 0-15 and 16-31 both cover N=0..15)
  const int mrow = lane & 15;             // M within tile for A fragment
  const int krow = (lane < 16) ? 0 : 2;   // A/B: lower lanes hold K=k0,k0+1; upper K=k0+2,k0+3
  for (int k0 = 0; k0 < Ktot; k0 += 4) {
    const int ka = k0 + krow;
    const int kb = ka + 1;
    v2f b;
    b.x = (ka < Kvalid) ? Bg[ka * 128 + n0 + col] : 0.f;
    b.y = (kb < Kvalid) ? Bg[kb * 128 + n0 + col] : 0.f;
#pragma unroll
    for (int m = 0; m < 4; ++m) {
      v2f a;
      a.x = As[(m * 16 + mrow) * lda + ka];   // padded A cols are zero for ka >= Kvalid
      a.y = As[(m * 16 + mrow) * lda + kb];
      acc[m] = wmma4(a, b, acc[m]);
    }
  }
  const int mOff = (lane < 16) ? 0 : 8;   // D: VGPR r holds M=r (lanes 0-15) / M=r+8 (lanes 16-31)
  const float bv = bias[n0 + col];
#pragma unroll
  for (int m = 0; m < 4; ++m) {
#pragma unroll
    for (int r = 0; r < 8; ++r) {
      float v = acc[m][r] + bv;
      if (relu) v = fmaxf(v, 0.f);
      Ds[(m * 16 + r + mOff) * 128 + (n0 + col)] = v;
    }
  }
}

__global__ __launch_bounds__(256) void k_mlp(const float* __restrict__ xbuf,
                                             const float* __restrict__ w1, const float* __restrict__ b1,
                                             const float* __restrict__ w2, const float* __restrict__ b2,
                                             const float* __restrict__ w3, const float* __restrict__ b3,
                                             const float* __restrict__ lnfs, const float* __restrict__ lnfb,
                                             const int* __restrict__ ripple, int S,
                                             float* __restrict__ out) {
  __shared__ float bufA[64 * 128];   // ping (x lives in first 64*20 floats initially)
  __shared__ float bufB[64 * 128];   // pong
  const int tid = threadIdx.x;
  for (int i = tid; i < 64 * 20; i += 256) bufA[i] = xbuf[i];
  __syncthreads();
  const int lane = tid & 31;
  const int n0 = (tid >> 5) * 16;    // 8 waves x 16 columns = 128
  gemm_tile(bufA, 20, 20, 17, w1, b1, bufB, true,  lane, n0);  // h1 = relu(x@w1 + b1)
  __syncthreads();
  gemm_tile(bufB, 128, 128, 128, w2, b2, bufA, true,  lane, n0); // h2 = relu(h1@w2 + b2)
  __syncthreads();
  gemm_tile(bufA, 128, 128, 128, w3, b3, bufB, false, lane, n0); // h3 = h2@w3 + b3
  __syncthreads();
  if (tid < S && tid < 64) {         // final layernorm + scatter-add into the copied output
    float mu = 0.f;
    for (int c = 0; c < 128; ++c) mu += bufB[tid * 128 + c];
    mu *= (1.f / 128.f);
    float var = 0.f;
    for (int c = 0; c < 128; ++c) { float d = bufB[tid * 128 + c] - mu; var += d * d; }
    var *= (1.f / 128.f);
    float inv = rsqrtf(var + 1e-5f);
    const long long row = (long long)ripple[tid] * 128;
    for (int c = 0; c < 128; ++c) {
      float h = (bufB[tid * 128 + c] - mu) * inv * lnfs[c] + lnfb[c];
      out[row + c] += h;
    }
  }
}

extern "C" void kernel_launch(void* const* d_in, const int* in_sizes, int n_in,
                              void* d_out, int out_size, void* d_ws, size_t ws_size,
                              hipStream_t stream) {
  const float* nf   = (const float*)d_in[0];
  const float* wpos = (const float*)d_in[1];
  const float* mpos = (const float*)d_in[2];
  const float* lat  = (const float*)d_in[3];
  const int*   rip  = (const int*)d_in[4];
  const float* ln0s = (const float*)d_in[5];
  const float* ln0b = (const float*)d_in[6];
  const float* w1   = (const float*)d_in[7];
  const float* b1   = (const float*)d_in[8];
  const float* w2   = (const float*)d_in[9];
  const float* b2   = (const float*)d_in[10];
  const float* w3   = (const float*)d_in[11];
  const float* b3   = (const float*)d_in[12];
  const float* lnfs = (const float*)d_in[13];
  const float* lnfb = (const float*)d_in[14];
  float* out = (float*)d_out;

  const int N = in_sizes[3] / 128;
  const int S = in_sizes[4];

  // workspace layout (u32 indices into d_ws)
  unsigned* hist0   = (unsigned*)d_ws;                // [0 .. 65535]
  unsigned* hist1   = hist0 + 65536;                  // [65536 .. 131071]
  unsigned* state   = hist1 + 65536;                  // [131072 .. 131075]
  unsigned* tiectr  = state + 4;                      // [131076]
  int*      tielist = (int*)(tiectr + 1);             // [131077 .. 131204]
  double*   partials = (double*)((char*)d_ws + 131208ull * 4ull);   // 8-byte aligned
  float*    xbuf     = (float*)((char*)partials + (size_t)RB * 8 * sizeof(double));

  k_zero_u32<<<520, 256, 0, stream>>>(hist0, 131077);                 // hists + state + tiectr
  k_hist0<<<1024, 256, 0, stream>>>(nf, N, hist0);
  k_scan<<<1, 256, 0, stream>>>(hist0, state, 0);
  k_hist1<<<1024, 256, 0, stream>>>(nf, N, state, hist1);
  k_scan<<<1, 256, 0, stream>>>(hist1, state, 1);
  k_reduce<<<RB, 256, 0, stream>>>(nf, wpos, mpos, N, state, partials, tiectr, tielist);
  k_finalize<<<1, 64, 0, stream>>>(nf, wpos, mpos, rip, S, ln0s, ln0b,
                                   state, partials, tiectr, tielist, xbuf);
  const int n4 = out_size / 4;
  k_copy<<<8192, 256, 0, stream>>>((const v4f*)lat, (v4f*)out, n4);
  k_mlp<<<1, 256, 0, stream>>>(xbuf, w1, b1, w2, b2, w3, b3, lnfs, lnfb, rip, S, out);
}